// FMEncoderCat_47751446397030
// MI455X (gfx1250) — compile-verified
//
#include <hip/hip_runtime.h>

// ---------------- problem constants (match reference) ----------------
#define PN       20000
#define DN       10000
#define RREL     64
#define EPER_DD  16384
#define EDD      (RREL * EPER_DD)   // 1048576
#define EPP      640000
#define EDP      160000
#define IND      2048               // IN_DRUG
#define INP      128                // IN_PROT
#define GH1      32
#define GH2      16
#define PDIM     16
#define NEMB     48
#define NH1      32
#define NH2      16
#define IN1      64                 // NEMB + PDIM

typedef __attribute__((ext_vector_type(2))) float v2f;
typedef __attribute__((ext_vector_type(8))) float v8f;

// D(16x16,f32) = A(16x4,f32) * B(4x16,f32) + C  — CDNA5 wave32 WMMA
__device__ __forceinline__ v8f wmma4(v2f a, v2f b, v8f c) {
  return __builtin_amdgcn_wmma_f32_16x16x4_f32(
      /*neg_a=*/false, a, /*neg_b=*/false, b,
      /*c_mod=*/(short)0, c, /*reuse_a=*/false, /*reuse_b=*/false);
}

__device__ __forceinline__ void atomAdd(float* p, float v) {
#ifdef __HIP_PLATFORM_AMD__
  unsafeAtomicAdd(p, v);   // native global_atomic_add_f32
#else
  atomicAdd(p, v);
#endif
}

// ---------------------------------------------------------------------
// Generic fp32 WMMA GEMM: C[M,N] = A[M,K] @ B[K,N]  (row-major, strides)
// One wave per 16x16 output tile; grid.x covers M tiles (4 waves/block),
// grid.y covers N tiles. Optional per-row divide epilogue (for /d_norm).
// M tiles may exceed grid coverage's multiple-of-4; guarded wave-uniformly
// so EXEC stays all-ones inside the WMMA region.
// ---------------------------------------------------------------------
__global__ void gemm_wmma(const float* __restrict__ A, const float* __restrict__ B,
                          float* __restrict__ C, int Mtiles, int K,
                          int lda, int ldb, int ldc,
                          const float* __restrict__ row_div) {
  int warp  = threadIdx.x >> 5;
  int mtile = blockIdx.x * (blockDim.x >> 5) + warp;
  if (mtile >= Mtiles) return;                 // wave-uniform
  int lane = threadIdx.x & 31;
  int r    = lane & 15;                        // M row / N col within tile
  int hi   = lane >> 4;                        // K half select
  const float* Arow = A + (size_t)(mtile * 16 + r) * lda;
  const float* Bcol = B + (size_t)blockIdx.y * 16 + r;
  v8f c = {0.f, 0.f, 0.f, 0.f, 0.f, 0.f, 0.f, 0.f};
  for (int k = 0; k < K; k += 4) {
    v2f a, b;
    a.x = Arow[k + 2 * hi];
    a.y = Arow[k + 2 * hi + 1];
    b.x = Bcol[(size_t)(k + 2 * hi) * ldb];
    b.y = Bcol[(size_t)(k + 2 * hi + 1) * ldb];
    c = wmma4(a, b, c);
  }
#pragma unroll
  for (int i = 0; i < 8; ++i) {
    int rr  = mtile * 16 + i + 8 * hi;
    float v = c[i];
    if (row_div) v /= row_div[rr];
    C[(size_t)rr * ldc + blockIdx.y * 16 + r] = v;
  }
}

// ---------------------------------------------------------------------
// RGCN edge messages via gather-WMMA + atomic scatter-mean numerator.
// 16-edge tile per wave; relation = tile_base / EPER_DD (EPER_DD%16==0 so
// a tile never straddles relations). msg[e] = X[src[e]] @ W[rel];
// numerator acc[dst[e]] += msg[e].
// ---------------------------------------------------------------------
__global__ void rgcn_edges_wmma(const float* __restrict__ X, const float* __restrict__ W,
                                const int* __restrict__ src, const int* __restrict__ dst,
                                float* __restrict__ acc, int IN, int OUT) {
  int warp = threadIdx.x >> 5;
  int tile = blockIdx.x * (blockDim.x >> 5) + warp;
  int base = tile * 16;
  int rel  = base >> 14;                       // / EPER_DD
  int lane = threadIdx.x & 31;
  int r    = lane & 15;
  int hi   = lane >> 4;
  const float* Wr   = W + (size_t)rel * IN * OUT + (size_t)blockIdx.y * 16 + r;
  const float* xrow = X + (size_t)src[base + r] * IN;   // gathered A row
  v8f c = {0.f, 0.f, 0.f, 0.f, 0.f, 0.f, 0.f, 0.f};
  for (int k = 0; k < IN; k += 4) {
    v2f a, b;
    a.x = xrow[k + 2 * hi];
    a.y = xrow[k + 2 * hi + 1];
    b.x = Wr[(size_t)(k + 2 * hi) * OUT];
    b.y = Wr[(size_t)(k + 2 * hi + 1) * OUT];
    c = wmma4(a, b, c);
  }
  int coln = blockIdx.y * 16 + r;
#pragma unroll
  for (int i = 0; i < 8; ++i) {
    int d = dst[base + i + 8 * hi];
    atomAdd(&acc[(size_t)d * OUT + coln], c[i]);
  }
}

// ------------------------- elementwise / scatter ----------------------
__global__ void k_count(const int* __restrict__ dst, float* __restrict__ cnt,
                        int n, int sub) {
  int i = blockIdx.x * blockDim.x + threadIdx.x;
  if (i < n) atomAdd(&cnt[dst[i] - sub], 1.0f);
}

__global__ void k_dinv(const float* __restrict__ cnt, float* __restrict__ dinv, int n) {
  int i = blockIdx.x * blockDim.x + threadIdx.x;
  if (i < n) dinv[i] = rsqrtf(cnt[i] + 1.0f);   // deg = segsum + self-loop
}

// agg[dst] += H[src] * dinv[src]*dinv[dst]   (one thread per edge-feature)
__global__ void k_gcn_scatter(const float* __restrict__ H, const int* __restrict__ src,
                              const int* __restrict__ dst, const float* __restrict__ dinv,
                              float* __restrict__ agg, int E, int logF) {
  long long i = (long long)blockIdx.x * blockDim.x + threadIdx.x;
  if (i >= ((long long)E << logF)) return;
  int e = (int)(i >> logF);
  int f = (int)(i & ((1 << logF) - 1));
  int s = src[e], d = dst[e];
  atomAdd(&agg[((size_t)d << logF) + f], H[((size_t)s << logF) + f] * dinv[s] * dinv[d]);
}

// agg = [relu]( agg + H*dinv^2 + bias )
__global__ void k_gcn_self(const float* __restrict__ H, const float* __restrict__ dinv,
                           const float* __restrict__ bias, float* __restrict__ agg,
                           int n, int logF, int relu) {
  int i = blockIdx.x * blockDim.x + threadIdx.x;
  if (i >= (n << logF)) return;
  int row = i >> logF;
  int f   = i & ((1 << logF) - 1);
  float v = agg[i] + H[i] * dinv[row] * dinv[row] + bias[f];
  agg[i] = relu ? fmaxf(v, 0.f) : v;
}

// prot->drug aggregation numerator (F = 16), dst ids are global [PN, PN+DN)
__global__ void k_dp_scatter(const float* __restrict__ H, const int* __restrict__ src,
                             const int* __restrict__ dst, float* __restrict__ agg, int E) {
  long long i = (long long)blockIdx.x * blockDim.x + threadIdx.x;
  if (i >= ((long long)E << 4)) return;
  int e = (int)(i >> 4);
  int f = (int)(i & 15);
  int d = dst[e] - PN;
  atomAdd(&agg[((size_t)d << 4) + f], H[((size_t)src[e] << 4) + f]);
}

__global__ void k_divcnt(float* __restrict__ x, const float* __restrict__ cnt,
                         int n, int logF) {
  int i = blockIdx.x * blockDim.x + threadIdx.x;
  if (i < n) x[i] /= fmaxf(cnt[i >> logF], 1.0f);
}

// out = [relu]( out + acc / max(cnt,1) )
__global__ void k_combine(float* __restrict__ out, const float* __restrict__ acc,
                          const float* __restrict__ cnt, int n, int logF, int relu) {
  int i = blockIdx.x * blockDim.x + threadIdx.x;
  if (i >= n) return;
  float v = out[i] + acc[i] / fmaxf(cnt[i >> logF], 1.0f);
  out[i] = relu ? fmaxf(v, 0.f) : v;
}

// ---------------------------------------------------------------------
extern "C" void kernel_launch(void* const* d_in, const int* in_sizes, int n_in,
                              void* d_out, int out_size, void* d_ws, size_t ws_size,
                              hipStream_t stream) {
  (void)in_sizes; (void)n_in; (void)out_size; (void)ws_size;
  const float* x_drug   = (const float*)d_in[0];
  const int*   dd_ei    = (const int*)  d_in[1];
  /* d_in[2] dd_edge_type, d_in[3] dd_range_list: implicit via sorted blocks */
  const float* d_norm   = (const float*)d_in[4];
  const float* x_prot   = (const float*)d_in[5];
  const int*   pp_ei    = (const int*)  d_in[6];
  const int*   dp_ei    = (const int*)  d_in[7];
  /* d_in[8] dp_range_list unused */
  const float* gcn_w1   = (const float*)d_in[9];
  const float* gcn_b1   = (const float*)d_in[10];
  const float* gcn_w2   = (const float*)d_in[11];
  const float* gcn_b2   = (const float*)d_in[12];
  const float* embed    = (const float*)d_in[13];
  const float* hgcn_w   = (const float*)d_in[14];
  const float* r1_basis = (const float*)d_in[15];   // [32,64,32] -> [32,2048]
  const float* r1_att   = (const float*)d_in[16];   // [64,32]
  const float* r1_root  = (const float*)d_in[17];   // [64,32]
  const float* r2_basis = (const float*)d_in[18];   // [32,32,16] -> [32,512]
  const float* r2_att   = (const float*)d_in[19];
  const float* r2_root  = (const float*)d_in[20];   // [32,16]

  // -------- workspace layout (floats) --------
  float* ws = (float*)d_ws;
  size_t off = 0;
  auto alloc = [&](size_t nf) { float* p = ws + off; off += nf; return p; };
  float* deg_pp  = alloc(PN);
  float* dinv_pp = alloc(PN);
  float* h1      = alloc((size_t)PN * GH1);
  float* h1agg   = alloc((size_t)PN * GH1);
  float* h2      = alloc((size_t)PN * GH2);
  float* h2agg   = alloc((size_t)PN * GH2);
  float* dp_cnt  = alloc(DN);
  float* hd_agg  = alloc((size_t)DN * GH2);
  float* dd_cnt  = alloc(DN);
  float* xd      = alloc((size_t)DN * IN1);
  float* xd1     = alloc((size_t)DN * NH1);
  float* acc1    = alloc((size_t)DN * NH1);
  float* acc2    = alloc((size_t)DN * NH2);
  float* w1      = alloc((size_t)RREL * IN1 * NH1);
  float* w2      = alloc((size_t)RREL * NH1 * NH2);
  hipMemsetAsync(d_ws, 0, off * sizeof(float), stream);   // zero accumulators

  const int* pp_src = pp_ei;  const int* pp_dst = pp_ei + EPP;
  const int* dp_src = dp_ei;  const int* dp_dst = dp_ei + EDP;
  const int* dd_src = dd_ei;  const int* dd_dst = dd_ei + EDD;
  float* out = (float*)d_out;

  const int T = 256;
  const int MT_P = PN / 16;   // 1250
  const int MT_D = DN / 16;   // 625
  auto mblk = [](int mt) { return (mt + 3) / 4; };   // 4 waves (128 thr) / block

  // --- protein GCN layer 1 ---
  k_count<<<(EPP + T - 1) / T, T, 0, stream>>>(pp_dst, deg_pp, EPP, 0);
  k_dinv<<<(PN + T - 1) / T, T, 0, stream>>>(deg_pp, dinv_pp, PN);
  gemm_wmma<<<dim3(mblk(MT_P), GH1 / 16), 128, 0, stream>>>(
      x_prot, gcn_w1, h1, MT_P, INP, INP, GH1, GH1, nullptr);
  {
    long long tot = (long long)EPP * GH1;
    k_gcn_scatter<<<(unsigned)((tot + T - 1) / T), T, 0, stream>>>(
        h1, pp_src, pp_dst, dinv_pp, h1agg, EPP, 5);
  }
  k_gcn_self<<<(PN * GH1 + T - 1) / T, T, 0, stream>>>(h1, dinv_pp, gcn_b1, h1agg, PN, 5, 1);

  // --- protein GCN layer 2 ---
  gemm_wmma<<<dim3(mblk(MT_P), GH2 / 16), 128, 0, stream>>>(
      h1agg, gcn_w2, h2, MT_P, GH1, GH1, GH2, GH2, nullptr);
  {
    long long tot = (long long)EPP * GH2;
    k_gcn_scatter<<<(unsigned)((tot + T - 1) / T), T, 0, stream>>>(
        h2, pp_src, pp_dst, dinv_pp, h2agg, EPP, 4);
  }
  k_gcn_self<<<(PN * GH2 + T - 1) / T, T, 0, stream>>>(h2, dinv_pp, gcn_b2, h2agg, PN, 4, 0);

  // --- hierarchy: prot -> drug mean, project to xd[:,48:64] ---
  k_count<<<(EDP + T - 1) / T, T, 0, stream>>>(dp_dst, dp_cnt, EDP, PN);
  {
    long long tot = (long long)EDP * GH2;
    k_dp_scatter<<<(unsigned)((tot + T - 1) / T), T, 0, stream>>>(
        h2agg, dp_src, dp_dst, hd_agg, EDP);
  }
  k_divcnt<<<(DN * GH2 + T - 1) / T, T, 0, stream>>>(hd_agg, dp_cnt, DN * GH2, 4);
  gemm_wmma<<<dim3(mblk(MT_D), PDIM / 16), 128, 0, stream>>>(
      hd_agg, hgcn_w, xd + NEMB, MT_D, GH2, GH2, PDIM, IN1, nullptr);

  // --- drug embedding into xd[:,0:48], fused /d_norm ---
  gemm_wmma<<<dim3(mblk(MT_D), NEMB / 16), 128, 0, stream>>>(
      x_drug, embed, xd, MT_D, IND, IND, NEMB, IN1, d_norm);

  // --- basis-decomposed relation weights: w = att @ basis ---
  gemm_wmma<<<dim3(1, (IN1 * NH1) / 16), 128, 0, stream>>>(
      r1_att, r1_basis, w1, RREL / 16, 32, 32, IN1 * NH1, IN1 * NH1, nullptr);
  gemm_wmma<<<dim3(1, (NH1 * NH2) / 16), 128, 0, stream>>>(
      r2_att, r2_basis, w2, RREL / 16, 32, 32, NH1 * NH2, NH1 * NH2, nullptr);

  // --- dd degree counts (mean over ALL relations' edges per dst) ---
  k_count<<<(EDD + T - 1) / T, T, 0, stream>>>(dd_dst, dd_cnt, EDD, 0);

  // --- RGCN layer 1: xd1 = relu(mean_agg + xd @ root1) ---
  gemm_wmma<<<dim3(mblk(MT_D), NH1 / 16), 128, 0, stream>>>(
      xd, r1_root, xd1, MT_D, IN1, IN1, NH1, NH1, nullptr);
  rgcn_edges_wmma<<<dim3(EDD / 16 / 4, NH1 / 16), 128, 0, stream>>>(
      xd, w1, dd_src, dd_dst, acc1, IN1, NH1);
  k_combine<<<(DN * NH1 + T - 1) / T, T, 0, stream>>>(xd1, acc1, dd_cnt, DN * NH1, 5, 1);

  // --- RGCN layer 2: out = mean_agg + xd1 @ root2 ---
  gemm_wmma<<<dim3(mblk(MT_D), NH2 / 16), 128, 0, stream>>>(
      xd1, r2_root, out, MT_D, NH1, NH1, NH2, NH2, nullptr);
  rgcn_edges_wmma<<<dim3(EDD / 16 / 4, NH2 / 16), 128, 0, stream>>>(
      xd1, w2, dd_src, dd_dst, acc2, NH1, NH2);
  k_combine<<<(DN * NH2 + T - 1) / T, T, 0, stream>>>(out, acc2, dd_cnt, DN * NH2, 4, 0);
}